// LiftSplat_62869731279372
// MI455X (gfx1250) — compile-verified
//
#include <hip/hip_runtime.h>
#include <hip/hip_bf16.h>

// ---------------- problem constants (from reference module) ----------------
#define N_CAM   6
#define C_DIM   160
#define H_DIM   16
#define W_DIM   44
#define D_DIM   112
#define NX_BINS 200
#define NY_BINS 200
#define HW      (H_DIM * W_DIM)          // 704
#define NBINS   (NX_BINS * NY_BINS)      // 40000
#define DEPTH_MIN 1.0f
#define DEPTH_MAX 57.0f
#define CT_PER_WAVE 5                    // 10 channel tiles split over 2 waves

typedef __attribute__((ext_vector_type(2))) float v2f;
typedef __attribute__((ext_vector_type(8))) float v8f;

// ---------------- kernel 0: zero the output ----------------
__global__ void LiftSplat_zero_kernel(float* __restrict__ out, int n) {
    int i = blockIdx.x * blockDim.x + threadIdx.x;
    if (i < n) out[i] = 0.0f;
}

// ---------------- kernel 1: per-camera M = R * inv(K), t = E[:3,3] ----------------
__global__ void LiftSplat_setup_kernel(const float* __restrict__ intr,
                                       const float* __restrict__ extr,
                                       float* __restrict__ camM,   // [N_CAM][9]
                                       float* __restrict__ camT) { // [N_CAM][3]
    int n = threadIdx.x;
    if (n >= N_CAM) return;
    const float* Kc = intr + n * 9;
    float k00=Kc[0],k01=Kc[1],k02=Kc[2],
          k10=Kc[3],k11=Kc[4],k12=Kc[5],
          k20=Kc[6],k21=Kc[7],k22=Kc[8];
    float c00 =  (k11*k22 - k12*k21);
    float c01 = -(k10*k22 - k12*k20);
    float c02 =  (k10*k21 - k11*k20);
    float det = k00*c00 + k01*c01 + k02*c02;
    float id  = 1.0f / det;
    float inv[3][3];
    inv[0][0] =  c00*id;
    inv[0][1] = -(k01*k22 - k02*k21)*id;
    inv[0][2] =  (k01*k12 - k02*k11)*id;
    inv[1][0] =  c01*id;
    inv[1][1] =  (k00*k22 - k02*k20)*id;
    inv[1][2] = -(k00*k12 - k02*k10)*id;
    inv[2][0] =  c02*id;
    inv[2][1] = -(k00*k21 - k01*k20)*id;
    inv[2][2] =  (k00*k11 - k01*k10)*id;
    const float* E = extr + n * 16;       // 4x4 row-major
    for (int r = 0; r < 3; ++r) {
        for (int c = 0; c < 3; ++c) {
            camM[n*9 + r*3 + c] = E[r*4+0]*inv[0][c] + E[r*4+1]*inv[1][c] + E[r*4+2]*inv[2][c];
        }
        camT[n*3 + r] = E[r*4+3];
    }
}

// ---------------- kernel 2: fused lift (WMMA outer product) + splat (L2 atomics) ----
// One block of 2 wave32s per pixel ray (n,h,w); wave w owns channel tiles
// [w*5, w*5+5). 6*704 = 4224 blocks -> 8448 concurrent waves.
__global__ __launch_bounds__(64)
void LiftSplat_main_kernel(const float* __restrict__ depth,   // (N,D,H,W)
                           const float* __restrict__ ctx,     // (N,C,H,W)
                           const float* __restrict__ camM,
                           const float* __restrict__ camT,
                           const int*   __restrict__ ihp,
                           const int*   __restrict__ iwp,
                           float* __restrict__ out) {         // (C, NX*NY)
    __shared__ float ctxLds[C_DIM];
    __shared__ int   binLds[D_DIM];
    __shared__ float wgtLds[D_DIM];
    __shared__ int   runBin[128];
    __shared__ float runW[128];
    __shared__ int   runCnt;

    const int blk  = blockIdx.x;
    const int n    = blk / HW;
    const int rem  = blk - n * HW;       // h*W + w
    const int h    = rem / W_DIM;
    const int w    = rem - h * W_DIM;
    const int tid  = threadIdx.x;
    const int lane = tid & 31;
    const int wv   = tid >> 5;           // wave id: 0 or 1

    const float sx = (float)(*iwp) / (float)W_DIM;
    const float sy = (float)(*ihp) / (float)H_DIM;
    const float xs = (float)w * sx;
    const float ys = (float)h * sy;

    const float m0 = camM[n*9+0], m1 = camM[n*9+1], m2 = camM[n*9+2];
    const float m3 = camM[n*9+3], m4 = camM[n*9+4], m5 = camM[n*9+5];
    const float t0 = camT[n*3+0], t1 = camT[n*3+1];

    // ---- load the 160-wide context row for this pixel into LDS ----
    #pragma unroll
    for (int j = 0; j < 3; ++j) {
        int c = j * 64 + tid;
        if (c < C_DIM) ctxLds[c] = ctx[(size_t)(n * C_DIM + c) * HW + rem];
    }

    // ---- geometry + binning for all 112 depth slices (thread-parallel) ----
    const float dstep = (DEPTH_MAX - DEPTH_MIN) / (float)(D_DIM - 1);
    #pragma unroll
    for (int j = 0; j < 2; ++j) {
        int d = j * 64 + tid;
        if (d < D_DIM) {
            float dv = DEPTH_MIN + dstep * (float)d;
            float px = xs * dv, py = ys * dv, pz = dv;
            float gx = m0*px + m1*py + m2*pz + t0;
            float gy = m3*px + m4*py + m5*pz + t1;
            // reference: astype(int32) truncates toward zero; C cast matches
            int xi = (int)((gx + 50.0f) * 2.0f);
            int yi = (int)((gy + 50.0f) * 2.0f);
            bool valid = (xi >= 0) & (xi < NX_BINS) & (yi >= 0) & (yi < NY_BINS);
            binLds[d] = valid ? (xi * NY_BINS + yi) : -1;
            wgtLds[d] = depth[(size_t)(n * D_DIM + d) * HW + rem];
        }
    }
    __syncthreads();

    // ---- run-length merge consecutive depth slices hitting the same cell ----
    if (tid == 0) {
        int k = 0, prev = -1; float acc = 0.0f;
        for (int d = 0; d < D_DIM; ++d) {
            int b = binLds[d];
            if (b < 0) {
                if (prev >= 0) { runBin[k] = prev; runW[k] = acc; ++k; prev = -1; }
                continue;
            }
            if (b == prev) { acc += wgtLds[d]; }
            else {
                if (prev >= 0) { runBin[k] = prev; runW[k] = acc; ++k; }
                prev = b; acc = wgtLds[d];
            }
        }
        if (prev >= 0) { runBin[k] = prev; runW[k] = acc; ++k; }
        runCnt = k;
    }
    __syncthreads();

    const int K  = runCnt;                 // uniform
    const int hi = lane >> 4;              // C/D row group select
    const int lo = lane & 15;              // C/D column index
    const int nt = (K + 15) >> 4;

    // ---- preload this wave's 5 B-fragment values (dt-invariant) ----
    float bReg[CT_PER_WAVE];
    #pragma unroll
    for (int i = 0; i < CT_PER_WAVE; ++i) {
        bReg[i] = (hi == 0) ? ctxLds[(wv * CT_PER_WAVE + i) * 16 + lo] : 0.0f;
    }
    // per-lane base offset into out for this wave's first channel tile
    const int colBase0 = (wv * (CT_PER_WAVE * 16) + lo) * NBINS;

    for (int dt = 0; dt < nt; ++dt) {
        // A fragment 16x4 f32: only K=0 populated -> lanes 0-15 of VGPR0
        int r0 = dt * 16 + lo;
        v2f a;
        a.x = (hi == 0 && r0 < K) ? runW[r0] : 0.0f;
        a.y = 0.0f;

        // hoist scatter targets: bins depend only on dt, not on channel tile
        int  binv[8];
        bool ok[8];
        #pragma unroll
        for (int v = 0; v < 8; ++v) {
            int r = dt * 16 + v + hi * 8;
            ok[v]   = (r < K);
            binv[v] = ok[v] ? runBin[r] : 0;
        }

        #pragma unroll
        for (int i = 0; i < CT_PER_WAVE; ++i) {
            v2f b; b.x = bReg[i]; b.y = 0.0f;
            v8f cz = {0.f,0.f,0.f,0.f,0.f,0.f,0.f,0.f};
            // D[m,n] = runW[m] * ctx[n]  (full f32 rank-1 outer product)
            v8f dmat = __builtin_amdgcn_wmma_f32_16x16x4_f32(
                false, a, false, b, (short)0, cz, false, false);
            const int colBase = colBase0 + i * 16 * NBINS;
            // C/D layout: element (m = v + 8*hi, n = lo) lives in dmat[v]
            #pragma unroll
            for (int v = 0; v < 8; ++v) {
                if (ok[v]) {
                    atomicAdd(&out[(size_t)(colBase + binv[v])], dmat[v]);
                }
            }
        }
    }
}

// ---------------- host launcher ----------------
extern "C" void kernel_launch(void* const* d_in, const int* in_sizes, int n_in,
                              void* d_out, int out_size, void* d_ws, size_t ws_size,
                              hipStream_t stream) {
    // inputs (setup_inputs order): 0: image_features (unused by reference)
    const float* depth = (const float*)d_in[1];   // (1,6,112,16,44)
    const float* ctx   = (const float*)d_in[2];   // (1,6,160,16,44)
    const float* intr  = (const float*)d_in[3];   // (1,6,3,3)
    const float* extr  = (const float*)d_in[4];   // (1,6,4,4)
    const int*   ihp   = (const int*)d_in[5];     // scalar 256
    const int*   iwp   = (const int*)d_in[6];     // scalar 704

    float* out  = (float*)d_out;                  // (1,160,200,200)
    float* camM = (float*)d_ws;                   // 54 floats
    float* camT = camM + 64;                      // 18 floats

    int zb = (out_size + 255) / 256;
    LiftSplat_zero_kernel<<<zb, 256, 0, stream>>>(out, out_size);
    LiftSplat_setup_kernel<<<1, 32, 0, stream>>>(intr, extr, camM, camT);
    LiftSplat_main_kernel<<<N_CAM * HW, 64, 0, stream>>>(depth, ctx, camM, camT, ihp, iwp, out);
}